// NovelKQRAttention_69406671503974
// MI455X (gfx1250) — compile-verified
//
#include <hip/hip_runtime.h>
#include <hip/hip_bf16.h>

// ---------------------------------------------------------------------------
// NovelKQRAttention, algebraically simplified (exact):
//   softmax_k(a_q + b_k) == softmax_k(b_k)   (shift invariance)
//   => attn independent of q => out spatially constant per (n, channel)
//   => final = x + gamma * (proj_w @ wv + proj_b)  broadcast over H*W
// Remaining heavy step is the 38 MB stream over x (HBM bound, ~1.6us).
// Small GEMMs use V_WMMA_F32_16X16X4_F32 (wave32, full f32 precision).
// ---------------------------------------------------------------------------

typedef float  v2f __attribute__((ext_vector_type(2)));
typedef float  v8f __attribute__((ext_vector_type(8)));

#define NB      2
#define C_CH    576
#define HH      64
#define WW      64
#define HEADS   9
#define QKD     64
#define HK      32
#define WK      32
#define KTOT    (HK * WK)        // 1024 key positions
#define SPAT    (HH * WW)        // 4096

// workspace layout (floats)
#define WS_KB    0                       // 16 x 576  (head-padded folded key)
#define WS_B     (WS_KB + 16 * C_CH)     // NB x 16 x 1024 logits
#define WS_ATTN  (WS_B + NB * 16 * KTOT) // NB x 9 x 1024 softmax
#define WS_XA    (WS_ATTN + NB * HEADS * KTOT) // NB x 576 x 16
#define WS_WV    (WS_XA + NB * C_CH * 16)      // NB x 576
#define WS_G     (WS_WV + NB * C_CH)           // NB x 576

__device__ __forceinline__ int kv_spatial(int k) {
    // key index k in [0,1024) -> offset inside one (H,W)=(64,64) channel plane
    int ky = (k >> 5) << 1;          // 2*(k/32)
    int kx = (k & 31) << 1;          // 2*(k%32)
    return ky * WW + kx;
}

// ---- 1) kb[h,c] = sum_e appr_bias[h*64+e] * key_w[h*64+e, c], zero-pad h>=9
__global__ void fold_kb_kernel(const float* __restrict__ key_w,
                               const float* __restrict__ appr_bias,
                               float* __restrict__ kb) {
    int t = blockIdx.x * blockDim.x + threadIdx.x;   // 16*576 threads
    if (t >= 16 * C_CH) return;
    int h = t / C_CH;
    int c = t - h * C_CH;
    float acc = 0.f;
    if (h < HEADS) {
        const float* bias = appr_bias + h * QKD;
        const float* kw   = key_w + (size_t)(h * QKD) * C_CH + c;
        #pragma unroll 8
        for (int e = 0; e < QKD; ++e)
            acc += bias[e] * kw[(size_t)e * C_CH];
    }
    kb[t] = acc;
}

// ---- 2) B[n, 16(h), 1024(k)] = KB(16x576) @ Xkv(576x1024)  via WMMA f32 16x16x4
__global__ void __launch_bounds__(32)
b_gemm_kernel(const float* __restrict__ x, const float* __restrict__ kb,
              float* __restrict__ bmat) {
    int blk  = blockIdx.x;          // NB * 64 blocks, one wave each
    int n    = blk >> 6;
    int kt   = blk & 63;            // 16-wide key tile
    int lane = threadIdx.x;
    int mrow = lane & 15;           // A row (head), C/D column
    int hi   = (lane >= 16) ? 2 : 0;

    int kglob = kt * 16 + (lane & 15);
    const float* xn = x + (size_t)n * C_CH * SPAT;
    int spat = kv_spatial(kglob);

    v8f c = {0.f, 0.f, 0.f, 0.f, 0.f, 0.f, 0.f, 0.f};
    for (int c0 = 0; c0 < C_CH; c0 += 4) {
        v2f a, b;
        const float* ka = kb + mrow * C_CH + c0 + hi;
        a.x = ka[0];
        a.y = ka[1];
        const float* xb = xn + (size_t)(c0 + hi) * SPAT + spat;
        b.x = xb[0];
        b.y = xb[SPAT];
        c = __builtin_amdgcn_wmma_f32_16x16x4_f32(false, a, false, b,
                                                  (short)0, c, false, false);
    }
    // C/D layout: VGPR r -> (M=r, N=lane) for lane<16; (M=r+8, N=lane-16)
    int Mbase = (lane >= 16) ? 8 : 0;
    float* brow = bmat + ((size_t)n * 16 + Mbase) * KTOT + kt * 16 + (lane & 15);
    #pragma unroll
    for (int r = 0; r < 8; ++r)
        brow[(size_t)r * KTOT] = c[r];
}

// ---- 3) attn[n,h,:] = softmax over the 1024 keys
__global__ void __launch_bounds__(256)
softmax_kernel(const float* __restrict__ bmat, float* __restrict__ attn) {
    __shared__ float red[256];
    int n = blockIdx.x / HEADS;
    int h = blockIdx.x % HEADS;
    const float* row = bmat + ((size_t)n * 16 + h) * KTOT;
    float*       out = attn + ((size_t)n * HEADS + h) * KTOT;
    int t = threadIdx.x;

    float v0 = row[t], v1 = row[t + 256], v2 = row[t + 512], v3 = row[t + 768];
    float m = fmaxf(fmaxf(v0, v1), fmaxf(v2, v3));
    red[t] = m;
    __syncthreads();
    for (int s = 128; s > 0; s >>= 1) {
        if (t < s) red[t] = fmaxf(red[t], red[t + s]);
        __syncthreads();
    }
    m = red[0];
    __syncthreads();

    float e0 = __expf(v0 - m), e1 = __expf(v1 - m);
    float e2 = __expf(v2 - m), e3 = __expf(v3 - m);
    red[t] = e0 + e1 + e2 + e3;
    __syncthreads();
    for (int s = 128; s > 0; s >>= 1) {
        if (t < s) red[t] += red[t + s];
        __syncthreads();
    }
    float inv = 1.0f / red[0];
    out[t]       = e0 * inv;
    out[t + 256] = e1 * inv;
    out[t + 512] = e2 * inv;
    out[t + 768] = e3 * inv;
}

// ---- 4) XA[n, 576(c), 16(h)] = Xkv(576x1024) @ AttnPad(1024x16)  via WMMA
__global__ void __launch_bounds__(32)
xa_gemm_kernel(const float* __restrict__ x, const float* __restrict__ attn,
               float* __restrict__ xa) {
    int blk  = blockIdx.x;          // NB * 36 blocks (channel tiles of 16)
    int n    = blk / 36;
    int mt   = blk % 36;
    int lane = threadIdx.x;
    int hi   = (lane >= 16) ? 2 : 0;

    int mglob = mt * 16 + (lane & 15);           // channel row for A
    int hcol  = lane & 15;                       // head column for B (pad >=9)
    const float* xn = x + ((size_t)n * C_CH + mglob) * SPAT;
    const float* an = attn + ((size_t)n * HEADS + ((hcol < HEADS) ? hcol : 0)) * KTOT;
    float hvalid = (hcol < HEADS) ? 1.0f : 0.0f;

    v8f c = {0.f, 0.f, 0.f, 0.f, 0.f, 0.f, 0.f, 0.f};
    for (int k0 = 0; k0 < KTOT; k0 += 4) {
        v2f a, b;
        a.x = xn[kv_spatial(k0 + hi)];
        a.y = xn[kv_spatial(k0 + hi + 1)];
        b.x = an[k0 + hi]     * hvalid;
        b.y = an[k0 + hi + 1] * hvalid;
        c = __builtin_amdgcn_wmma_f32_16x16x4_f32(false, a, false, b,
                                                  (short)0, c, false, false);
    }
    int Mbase = (lane >= 16) ? 8 : 0;
    float* xrow = xa + ((size_t)n * C_CH + mt * 16 + Mbase) * 16 + (lane & 15);
    #pragma unroll
    for (int r = 0; r < 8; ++r)
        xrow[(size_t)r * 16] = c[r];
}

// ---- 5a) wv[n,o] = sum_c value_w[o,c] * xa[n,c,h(o)]
__global__ void wv_kernel(const float* __restrict__ value_w,
                          const float* __restrict__ xa,
                          float* __restrict__ wv) {
    int t = blockIdx.x * blockDim.x + threadIdx.x;   // NB*576
    if (t >= NB * C_CH) return;
    int n = t / C_CH;
    int o = t - n * C_CH;
    int h = o >> 6;
    const float* vw = value_w + (size_t)o * C_CH;
    const float* xb = xa + (size_t)n * C_CH * 16 + h;
    float acc = 0.f;
    #pragma unroll 8
    for (int c = 0; c < C_CH; ++c)
        acc += vw[c] * xb[c * 16];
    wv[t] = acc;
}

// ---- 5b) g[n,o] = gamma * (sum_c proj_w[o,c]*wv[n,c] + proj_b[o])
__global__ void proj_kernel(const float* __restrict__ proj_w,
                            const float* __restrict__ proj_b,
                            const float* __restrict__ gamma,
                            const float* __restrict__ wv,
                            float* __restrict__ g) {
    int t = blockIdx.x * blockDim.x + threadIdx.x;   // NB*576
    if (t >= NB * C_CH) return;
    int n = t / C_CH;
    int o = t - n * C_CH;
    const float* pw = proj_w + (size_t)o * C_CH;
    const float* wn = wv + (size_t)n * C_CH;
    float acc = proj_b[o];
    #pragma unroll 8
    for (int c = 0; c < C_CH; ++c)
        acc += pw[c] * wn[c];
    g[t] = gamma[0] * acc;
}

// ---- 6) out[n,c,y,x] = x[n,c,y,x] + g[n,c]   (HBM-bound stream, float4)
__global__ void __launch_bounds__(256)
residual_kernel(const float* __restrict__ x, const float* __restrict__ g,
                float* __restrict__ out) {
    size_t i4 = (size_t)blockIdx.x * blockDim.x + threadIdx.x;
    size_t total4 = (size_t)NB * C_CH * SPAT / 4;
    if (i4 >= total4) return;
    size_t e  = i4 * 4;
    int ch = (int)((e >> 12) % C_CH);        // / 4096
    int n  = (int)(e / ((size_t)C_CH * SPAT));
    __builtin_prefetch(x + e + 8192, 0, 1);  // gfx1250 global_prefetch hint
    float4 xv = *(const float4*)(x + e);
    float  gv = g[n * C_CH + ch];
    float4 ov = {xv.x + gv, xv.y + gv, xv.z + gv, xv.w + gv};
    *(float4*)(out + e) = ov;
}

extern "C" void kernel_launch(void* const* d_in, const int* in_sizes, int n_in,
                              void* d_out, int out_size, void* d_ws, size_t ws_size,
                              hipStream_t stream) {
    const float* x         = (const float*)d_in[0];
    const float* key_w     = (const float*)d_in[1];
    // d_in[2] offset_w, d_in[3] dconv_w, d_in[8] appr_bias_q: eliminated exactly
    const float* value_w   = (const float*)d_in[4];
    const float* proj_w    = (const float*)d_in[5];
    const float* proj_b    = (const float*)d_in[6];
    const float* appr_bias = (const float*)d_in[7];
    const float* gamma     = (const float*)d_in[9];
    float* out = (float*)d_out;
    float* ws  = (float*)d_ws;

    float* kb   = ws + WS_KB;
    float* bmat = ws + WS_B;
    float* attn = ws + WS_ATTN;
    float* xa   = ws + WS_XA;
    float* wv   = ws + WS_WV;
    float* g    = ws + WS_G;

    fold_kb_kernel<<<(16 * C_CH + 255) / 256, 256, 0, stream>>>(key_w, appr_bias, kb);
    b_gemm_kernel<<<NB * 64, 32, 0, stream>>>(x, kb, bmat);
    softmax_kernel<<<NB * HEADS, 256, 0, stream>>>(bmat, attn);
    xa_gemm_kernel<<<NB * 36, 32, 0, stream>>>(x, attn, xa);
    wv_kernel<<<(NB * C_CH + 255) / 256, 256, 0, stream>>>(value_w, xa, wv);
    proj_kernel<<<(NB * C_CH + 255) / 256, 256, 0, stream>>>(proj_w, proj_b, gamma, wv, g);
    size_t total4 = (size_t)NB * C_CH * SPAT / 4;
    residual_kernel<<<(unsigned)((total4 + 255) / 256), 256, 0, stream>>>(x, g, out);
}